// STEM_73151882986190
// MI455X (gfx1250) — compile-verified
//
#include <hip/hip_runtime.h>
#include <hip/hip_bf16.h>
#include <math.h>

// ---------------- problem constants ----------------
constexpr int B  = 2;
constexpr int C  = 192;
constexpr int D  = 32;
constexpr int H  = 48;
constexpr int W  = 48;
constexpr int HW = H * W;                // 2304
constexpr int S  = D * H * W;            // 73728 spatial per (b,c)
constexpr long long NTOT = (long long)B * C * S;  // 28,311,552

// conv tile (pass 2): left w-halo padded to 4 so valid data is 16B-aligned
constexpr int TD = 4;                    // d-tile
constexpr int TH = 8;                    // h-tile
constexpr int LD = TD + 6;               // 10 (halo 3 each side in d)
constexpr int LH = TH + 6;               // 14 (halo 3 each side in h)
constexpr int LW = 56;                   // 4 (pad) + 48 + 4 (pad) — 224B row stride
constexpr int LSZ = LD * LH * LW;        // 7840 floats = 31.4 KB LDS
constexpr int ROWS = LD * LH;            // 140 rows
constexpr int CPR  = LW / 4;             // 14 16-byte chunks per row
constexpr int TILES_H = H / TH;          // 6
constexpr int TILES_D = D / TD;          // 8
constexpr int TILES   = TILES_H * TILES_D; // 48 tiles per (b,c) plane

typedef int v4i __attribute__((ext_vector_type(4)));
typedef __attribute__((address_space(1))) v4i* gptr_v4i;
typedef __attribute__((address_space(3))) v4i* lptr_v4i;

// ---------------- CDNA5 async global->LDS helpers ----------------
__device__ __forceinline__ void async_g2l_b128(const float* g, float* l) {
#if __has_builtin(__builtin_amdgcn_global_load_async_to_lds_b128)
  __builtin_amdgcn_global_load_async_to_lds_b128(
      (gptr_v4i)(g), (lptr_v4i)(l), /*offset=*/0, /*cpol=*/0);
#else
  asm volatile("global_load_async_to_lds_b128 %0, %1, off"
               :: "v"((unsigned)(size_t)l), "v"(g)
               : "memory");
#endif
}

__device__ __forceinline__ void wait_asynccnt0() {
#if __has_builtin(__builtin_amdgcn_s_wait_asynccnt)
  __builtin_amdgcn_s_wait_asynccnt(0);
#else
  asm volatile("s_wait_asynccnt 0" ::: "memory");
#endif
}

// =====================================================================
// Pass 1: channel LayerNorm + dual per-channel affine + SiLU
//   a = silu(LN(xs)*qk_sp + xt*qk_tm),  v = LN(xs)*v_sp + xt*v_tm
// one thread per voxel; channel loop has uniform c -> weight reads are s_loads
// =====================================================================
__global__ __launch_bounds__(256) void pass1_ln_affine_silu(
    const float* __restrict__ xs, const float* __restrict__ xt,
    const float* __restrict__ ln_w, const float* __restrict__ ln_b,
    const float* __restrict__ qk_sp_w, const float* __restrict__ qk_sp_b,
    const float* __restrict__ v_sp_w,  const float* __restrict__ v_sp_b,
    const float* __restrict__ qk_tm_w, const float* __restrict__ qk_tm_b,
    const float* __restrict__ v_tm_w,  const float* __restrict__ v_tm_b,
    float* __restrict__ abuf, float* __restrict__ vbuf)
{
  const int vi = blockIdx.x * 256 + threadIdx.x;   // 0 .. B*S-1
  const int b  = vi / S;
  const int s  = vi - b * S;
  const size_t base = (size_t)b * C * S + s;

  float sum = 0.f, sumsq = 0.f;
  for (int c = 0; c < C; ++c) {
    float x = xs[base + (size_t)c * S];
    sum   += x;
    sumsq  = fmaf(x, x, sumsq);
    // warm xt into near caches for the second loop (global_prefetch_b8, WGP scope)
    __builtin_prefetch(xt + base + (size_t)c * S, 0, 3);
  }
  const float mean = sum * (1.f / C);
  const float var  = sumsq * (1.f / C) - mean * mean;
  const float rstd = rsqrtf(var + 1e-6f);

  for (int c = 0; c < C; ++c) {
    const size_t idx = base + (size_t)c * S;
    const float x  = xs[idx];
    const float t  = xt[idx];
    const float xn = (x - mean) * rstd * ln_w[c] + ln_b[c];
    const float qk = fmaf(xn, qk_sp_w[c], qk_sp_b[c]) + fmaf(t, qk_tm_w[c], qk_tm_b[c]);
    const float vv = fmaf(xn, v_sp_w[c],  v_sp_b[c])  + fmaf(t, v_tm_w[c],  v_tm_b[c]);
    const float aa = qk * (1.f / (1.f + __expf(-qk)));   // SiLU
    abuf[idx] = aa;
    vbuf[idx] = vv;
  }
}

// =====================================================================
// Pass 2: depthwise 4x4x4 conv (dilation 2, pad 3) on `a`, times v,
//         staged x -> d_out, plus per-tile sum(x^2) partials for GRN.
// One block per (b, c, tile); halo tile staged with ASYNC b128 loads.
// =====================================================================
__global__ __launch_bounds__(256) void pass2_conv(
    const float* __restrict__ abuf, const float* __restrict__ vbuf,
    const float* __restrict__ attn_w, const float* __restrict__ attn_b,
    float* __restrict__ xstage, float* __restrict__ partials)
{
  __shared__ __align__(16) float tile[LSZ];
  __shared__ float red[256];

  const int bid = blockIdx.x;
  const int t   = bid % TILES;
  const int c   = (bid / TILES) % C;
  const int b   = bid / (TILES * C);
  const int h0  = (t % TILES_H) * TH;
  const int d0  = (t / TILES_H) * TD;
  const int tid = threadIdx.x;

  const float* __restrict__ aplane = abuf + (size_t)(b * C + c) * S;

  // Stage halo tile in 16B chunks. Row layout (w): [4 pad][48 valid][4 pad].
  // Valid chunks (col4 1..12) -> async global->LDS b128 (ASYNCcnt);
  // OOB chunks -> ds_store_b128 of zeros. One writer per LDS chunk.
  for (int i = tid; i < ROWS * CPR; i += 256) {
    const int col4 = i % CPR;
    const int row  = i / CPR;
    const int lh = row % LH;
    const int ld = row / LH;
    const int gd = d0 + ld - 3;
    const int gh = h0 + lh - 3;
    float* ldst = &tile[row * LW + col4 * 4];
    const bool rowok = (unsigned)gd < (unsigned)D && (unsigned)gh < (unsigned)H;
    if (rowok && col4 >= 1 && col4 <= 12) {
      // global chunk: gw = (col4-1)*4 .. +3, 16B-aligned in both spaces
      async_g2l_b128(aplane + (size_t)gd * HW + gh * W + (col4 - 1) * 4, ldst);
    } else {
      *(float4*)ldst = make_float4(0.f, 0.f, 0.f, 0.f);
    }
  }
  wait_asynccnt0();
  __syncthreads();

  // Per-channel filter: c is block-uniform -> scalar loads
  float wgt[64];
#pragma unroll
  for (int k = 0; k < 64; ++k) wgt[k] = attn_w[c * 64 + k];
  const float ab = attn_b[c];

  float ssq = 0.f;
  for (int rr = 0; rr < (TD * TH * W) / 256; ++rr) {   // 6 outputs/thread
    const int j  = tid + 256 * rr;
    const int ow = j % W;
    const int q  = j / W;
    const int oh = q % TH;
    const int od = q / TH;
    float acc = 0.f;
    // input local coords: ld = od + 2kd, lh = oh + 2kh, lw = ow + 2kw + 1
    // (gw = ow - 3 + 2kw, left pad is 4)
#pragma unroll
    for (int kd = 0; kd < 4; ++kd)
#pragma unroll
      for (int kh = 0; kh < 4; ++kh)
#pragma unroll
        for (int kw = 0; kw < 4; ++kw)
          acc = fmaf(tile[(od + 2 * kd) * (LH * LW) + (oh + 2 * kh) * LW +
                          (ow + 2 * kw + 1)],
                     wgt[kd * 16 + kh * 4 + kw], acc);
    const size_t gidx =
        ((size_t)(b * C + c) * D + (d0 + od)) * HW + (h0 + oh) * W + ow;
    const float xv = (acc + ab) * vbuf[gidx];
    xstage[gidx] = xv;
    ssq = fmaf(xv, xv, ssq);
  }

  // deterministic block reduction of sum(x^2)
  red[tid] = ssq;
  __syncthreads();
  for (int ofs = 128; ofs > 0; ofs >>= 1) {
    if (tid < ofs) red[tid] += red[tid + ofs];
    __syncthreads();
  }
  if (tid == 0) partials[(size_t)(b * C + c) * TILES + t] = red[0];
}

// =====================================================================
// Pass 2b: gx[b,c] = sqrt(sum of partials); nx = gx / (mean_c gx + eps)
// one block per batch, 192 threads (one per channel), fixed-order sums.
// =====================================================================
__global__ __launch_bounds__(192) void pass2b_grn_stats(
    const float* __restrict__ partials, float* __restrict__ nxbuf)
{
  const int b = blockIdx.x;
  const int c = threadIdx.x;
  float s = 0.f;
  for (int t = 0; t < TILES; ++t)
    s += partials[(size_t)(b * C + c) * TILES + t];
  const float gx = sqrtf(s);

  __shared__ float sm[C];
  sm[c] = gx;
  __syncthreads();
  if (c == 0) {
    float m = 0.f;
    for (int i = 0; i < C; ++i) m += sm[i];
    sm[0] = m * (1.f / C);
  }
  __syncthreads();
  nxbuf[b * C + c] = gx / (sm[0] + 1e-6f);
}

// =====================================================================
// Pass 3: GRN apply + proj + residual, in place on d_out (which holds x)
//   out = xs + (gamma*(x*nx) + beta + x) * proj_w + proj_b
// float4 vectorized; S % 1024 == 0 -> each block within one (b,c):
// channel params are scalar loads.
// =====================================================================
__global__ __launch_bounds__(256) void pass3_finish(
    const float* __restrict__ xs, const float* __restrict__ nxbuf,
    const float* __restrict__ grn_gamma, const float* __restrict__ grn_beta,
    const float* __restrict__ proj_w, const float* __restrict__ proj_b,
    float* __restrict__ out)
{
  const size_t e4 = ((size_t)blockIdx.x * 256 + threadIdx.x) * 4;
  const int   bc  = (int)(e4 / S);        // block-uniform
  const int   c   = bc % C;
  const float nx  = nxbuf[bc];
  const float g   = grn_gamma[c];
  const float be  = grn_beta[c];
  const float pw  = proj_w[c];
  const float pb  = proj_b[c];

  float4 x  = *(const float4*)(out + e4);
  float4 x0 = *(const float4*)(xs + e4);
  float4 r;
  {
    float xg;
    xg  = fmaf(g, x.x * nx, be) + x.x;  r.x = x0.x + fmaf(xg, pw, pb);
    xg  = fmaf(g, x.y * nx, be) + x.y;  r.y = x0.y + fmaf(xg, pw, pb);
    xg  = fmaf(g, x.z * nx, be) + x.z;  r.z = x0.z + fmaf(xg, pw, pb);
    xg  = fmaf(g, x.w * nx, be) + x.w;  r.w = x0.w + fmaf(xg, pw, pb);
  }
  *(float4*)(out + e4) = r;
}

// =====================================================================
extern "C" void kernel_launch(void* const* d_in, const int* in_sizes, int n_in,
                              void* d_out, int out_size, void* d_ws, size_t ws_size,
                              hipStream_t stream) {
  const float* xs      = (const float*)d_in[0];
  const float* xt      = (const float*)d_in[1];
  const float* ln_w    = (const float*)d_in[2];
  const float* ln_b    = (const float*)d_in[3];
  const float* qk_sp_w = (const float*)d_in[4];
  const float* qk_sp_b = (const float*)d_in[5];
  const float* v_sp_w  = (const float*)d_in[6];
  const float* v_sp_b  = (const float*)d_in[7];
  const float* qk_tm_w = (const float*)d_in[8];
  const float* qk_tm_b = (const float*)d_in[9];
  const float* v_tm_w  = (const float*)d_in[10];
  const float* v_tm_b  = (const float*)d_in[11];
  const float* attn_w  = (const float*)d_in[12];   // [C,1,4,4,4]
  const float* attn_b  = (const float*)d_in[13];
  const float* grn_g   = (const float*)d_in[14];   // [1,C,1,1,1] -> C floats
  const float* grn_be  = (const float*)d_in[15];
  const float* proj_w  = (const float*)d_in[16];
  const float* proj_b  = (const float*)d_in[17];
  float* out = (float*)d_out;

  // workspace layout (floats): a | v | partials | nx
  float* wsf      = (float*)d_ws;
  float* abuf     = wsf;
  float* vbuf     = wsf + NTOT;
  float* partials = wsf + 2 * NTOT;                       // B*C*TILES = 18432
  float* nxbuf    = partials + (size_t)B * C * TILES;     // B*C = 384

  // pass 1: 147456 voxels
  pass1_ln_affine_silu<<<(B * S) / 256, 256, 0, stream>>>(
      xs, xt, ln_w, ln_b, qk_sp_w, qk_sp_b, v_sp_w, v_sp_b,
      qk_tm_w, qk_tm_b, v_tm_w, v_tm_b, abuf, vbuf);

  // pass 2: one block per (b,c,tile)
  pass2_conv<<<B * C * TILES, 256, 0, stream>>>(
      abuf, vbuf, attn_w, attn_b, out, partials);

  // pass 2b: GRN statistics
  pass2b_grn_stats<<<B, C, 0, stream>>>(partials, nxbuf);

  // pass 3: finish in place on d_out, float4 per thread
  pass3_finish<<<(unsigned)(NTOT / 1024), 256, 0, stream>>>(
      xs, nxbuf, grn_g, grn_be, proj_w, proj_b, out);
}